// FilteredCOMACritic_42923903156248
// MI455X (gfx1250) — compile-verified
//
#include <hip/hip_runtime.h>
#include <hip/hip_bf16.h>

// ---------------- problem constants ----------------
#define BB    8
#define TT    32
#define NAG   50
#define MTASK 100
#define MTOP  10
#define NNB   10
#define LL    5
#define HH    512
#define ROWS  (BB*TT*NAG)        // 12800
#define BT    (BB*TT)            // 256
#define IN_DIM  1610
#define IN_PAD  1664             // multiple of 32 (and 128)
#define OUT_DIM 11

typedef __attribute__((ext_vector_type(2))) float v2f;
typedef __attribute__((ext_vector_type(8))) float v8f;

// ---------------- workspace layout (floats) ----------------
#define WS_TB    0
#define WS_TASKS (WS_TB    + (size_t)ROWS*MTASK)
#define WS_BEST  (WS_TASKS + (size_t)ROWS*MTOP)
#define WS_NEIGH (WS_BEST  + (size_t)BT*NAG*NAG)
#define WS_W1P   (WS_NEIGH + (size_t)ROWS*NNB)
#define WS_FEAT  (WS_W1P   + (size_t)IN_PAD*HH)
#define WS_X1    (WS_FEAT  + (size_t)ROWS*IN_PAD)
#define WS_X2    (WS_X1    + (size_t)ROWS*HH)

// ---------------- kernel 1: total_beta = sum over L ----------------
__global__ void k_total_beta(const float* __restrict__ beta, float* __restrict__ tb) {
    int idx = blockIdx.x * blockDim.x + threadIdx.x;      // over ROWS*MTASK
    if (idx >= ROWS * MTASK) return;
    const float* p = beta + (size_t)idx * LL;
    float s = 0.f;
#pragma unroll
    for (int l = 0; l < LL; ++l) s += p[l];
    tb[idx] = s;
}

// ---------------- kernel 2: top-10 tasks per agent ----------------
__global__ void k_topk_tasks(const float* __restrict__ tb, int* __restrict__ tasks) {
    int gid = blockIdx.x * blockDim.x + threadIdx.x;      // over ROWS
    if (gid >= ROWS) return;
    const float* row = tb + (size_t)gid * MTASK;
    unsigned long long used0 = 0ull, used1 = 0ull;
#pragma unroll 1
    for (int k = 0; k < MTOP; ++k) {
        float bv = -__builtin_inff();
        int bi = -1;
        for (int m = 0; m < MTASK; ++m) {
            bool used = (m < 64) ? ((used0 >> m) & 1ull) : ((used1 >> (m - 64)) & 1ull);
            float v = row[m];
            if (!used && v > bv) { bv = v; bi = m; }
        }
        if (bi < 64) used0 |= 1ull << bi; else used1 |= 1ull << (bi - 64);
        tasks[gid * MTOP + k] = bi;
    }
}

// ---------------- kernel 3: best[bt,i,a] ----------------
__global__ void k_best(const float* __restrict__ tb, const int* __restrict__ tasks,
                       float* __restrict__ best) {
    int gid = blockIdx.x * blockDim.x + threadIdx.x;      // over BT*NAG*NAG
    if (gid >= BT * NAG * NAG) return;
    int a  = gid % NAG;
    int i  = (gid / NAG) % NAG;
    int bt = gid / (NAG * NAG);
    float m = -__builtin_inff();
    if (i != a) {
        const int*   trow = tasks + (size_t)(bt * NAG + i) * MTOP;
        const float* arow = tb    + (size_t)(bt * NAG + a) * MTASK;
#pragma unroll
        for (int k = 0; k < MTOP; ++k) {
            float v = arow[trow[k]];
            m = fmaxf(m, v);
        }
    }
    best[gid] = m;
}

// ---------------- kernel 4: top-10 neighbors per agent ----------------
__global__ void k_topk_nb(const float* __restrict__ best, int* __restrict__ neigh) {
    int gid = blockIdx.x * blockDim.x + threadIdx.x;      // over ROWS (== BT*NAG)
    if (gid >= ROWS) return;
    const float* row = best + (size_t)gid * NAG;
    unsigned long long used = 0ull;
#pragma unroll 1
    for (int k = 0; k < NNB; ++k) {
        float bv = -__builtin_inff();
        int bi = -1;
        for (int a = 0; a < NAG; ++a) {
            float v = row[a];
            if (!((used >> a) & 1ull) && v > bv) { bv = v; bi = a; }
        }
        if (bi < 0) bi = 0;
        used |= 1ull << bi;
        neigh[gid * NNB + k] = bi;
    }
}

// ---------------- kernel 5: pad W1 into workspace ----------------
__global__ void k_w1pad(const float* __restrict__ W1, float* __restrict__ W1p) {
    int idx = blockIdx.x * blockDim.x + threadIdx.x;      // over IN_PAD*HH
    if (idx >= IN_PAD * HH) return;
    int k = idx / HH;
    W1p[idx] = (k < IN_DIM) ? W1[idx] : 0.f;
}

// ---------------- kernel 6: build feature rows ----------------
__global__ void k_features(const float* __restrict__ beta,
                           const float* __restrict__ actions,
                           const float* __restrict__ power,
                           const int*   __restrict__ prev,
                           const int*   __restrict__ tasks,
                           const int*   __restrict__ neigh,
                           float* __restrict__ feat) {
    int row = blockIdx.x;                                 // over ROWS
    int bt  = row / NAG;
    __shared__ int s_tasks[MTOP];
    __shared__ int s_nb[NNB];
    int tid = threadIdx.x;
    if (tid < MTOP)                 s_tasks[tid] = tasks[row * MTOP + tid];
    else if (tid < MTOP + NNB)      s_nb[tid - MTOP] = neigh[row * NNB + (tid - MTOP)];
    __syncthreads();

    float* out = feat + (size_t)row * IN_PAD;
    for (int f = tid; f < IN_PAD; f += blockDim.x) {
        float val;
        if (f < 500) {                       // beta features, (M,N,L) order
            int k = f / (NNB * LL);
            int j = (f / LL) % NNB;
            int l = f % LL;
            int nb = s_nb[j];
            val = beta[(((size_t)bt * NAG + nb) * MTASK + s_tasks[k]) * LL + l];
        } else if (f < 1500) {               // actions_onehot features, (N,m) order
            int ff = f - 500;
            int j = ff / MTASK;
            int m = ff % MTASK;
            val = actions[((size_t)bt * NAG + s_nb[j]) * MTASK + m];
        } else if (f < 1510) {               // power features
            int j = f - 1500;
            val = power[(size_t)bt * NAG + s_nb[j]];
        } else if (f < IN_DIM) {             // prev-assign one-hot match, (M,N) order
            int ff = f - 1510;
            int k = ff / NNB;
            int j = ff % NNB;
            val = (s_tasks[k] == prev[(size_t)bt * NAG + s_nb[j]]) ? 1.f : 0.f;
        } else {
            val = 0.f;                       // zero pad to IN_PAD
        }
        out[f] = val;
    }
}

// ---------------- kernel 7: generic fp32 WMMA GEMM (+bias, +relu) ----------------
// C[M,N] = A[M,K] * B[K,N] + bias
// Block tile 128x128, 8 waves (4 along M x 2 along N), each wave 32x64 = 2x4 WMMA tiles.
#define GBM 128
#define GBN 128
#define GBK 32
#define ALDS 36   // GBK+4 : 144B rows -> b128-aligned staging, 8B-aligned v2f frag reads,
                  // and conflict-free frag access (36*r mod 64 distinct for r=0..15)

__global__ __launch_bounds__(256) void k_gemm_wmma(
    const float* __restrict__ A, int lda,
    const float* __restrict__ B, int ldb,
    const float* __restrict__ bias,
    float* __restrict__ C, int ldc,
    int M, int N, int K, int relu)
{
    __shared__ float As[GBM * ALDS];   // 18432 B
    __shared__ float Bs[GBK * GBN];    // 16384 B

    const int tid  = threadIdx.x;
    const int lane = tid & 31;
    const int wave = tid >> 5;         // 0..7
    const int wm   = wave & 3;         // 4 waves along M (32 rows each)
    const int wn   = wave >> 2;        // 2 waves along N (64 cols each)
    const int blockM = blockIdx.y * GBM;
    const int blockN = blockIdx.x * GBN;

    const int r16   = lane & 15;
    const int khalf = (lane >> 4) << 1;   // 0 or 2

    v8f acc[2][4] = {};

    for (int k0 = 0; k0 < K; k0 += GBK) {
        // ---- stage A tile (128x32), b128 global -> b128 LDS ----
#pragma unroll
        for (int i = 0; i < 4; ++i) {
            int g  = tid + i * 256;
            int ar = g >> 3;                 // row in tile
            int ac = (g & 7) << 2;           // col (float4 group)
            float4 v = make_float4(0.f, 0.f, 0.f, 0.f);
            int grow = blockM + ar;
            if (grow < M)
                v = *(const float4*)&A[(size_t)grow * lda + k0 + ac];
            *(float4*)&As[ar * ALDS + ac] = v;
        }
        // ---- stage B tile (32x128) ----
        if (blockN + GBN <= N) {
#pragma unroll
            for (int i = 0; i < 4; ++i) {
                int g  = tid + i * 256;
                int br = g >> 5;
                int bc = (g & 31) << 2;
                *(float4*)&Bs[br * GBN + bc] =
                    *(const float4*)&B[(size_t)(k0 + br) * ldb + blockN + bc];
            }
        } else {
#pragma unroll
            for (int i = 0; i < 4; ++i) {
                int g  = tid + i * 256;
                int br = g >> 5;
                int bc = (g & 31) << 2;
#pragma unroll
                for (int c = 0; c < 4; ++c) {
                    int col = blockN + bc + c;
                    Bs[br * GBN + bc + c] =
                        (col < N) ? B[(size_t)(k0 + br) * ldb + col] : 0.f;
                }
            }
        }
        // ---- prefetch next K-chunk while this one is consumed ----
        if (k0 + GBK < K) {
            {
                int g  = tid;
                int ar = g >> 3;
                int ac = (g & 7) << 2;
                int grow = blockM + ar;
                if (grow < M)
                    __builtin_prefetch(&A[(size_t)grow * lda + k0 + GBK + ac], 0, 3);
            }
            {
                int g  = tid;
                int br = g >> 5;
                int bc = (g & 31) << 2;
                if (blockN + bc < N)
                    __builtin_prefetch(&B[(size_t)(k0 + GBK + br) * ldb + blockN + bc], 0, 3);
            }
        }
        __syncthreads();

        // ---- 8 K-steps of v_wmma_f32_16x16x4_f32 (8 WMMAs each) ----
#pragma unroll
        for (int kk = 0; kk < GBK; kk += 4) {
            v2f afrag[2];
            v2f bfrag[4];
#pragma unroll
            for (int mi = 0; mi < 2; ++mi) {
                const float* p = &As[(wm * 32 + mi * 16 + r16) * ALDS + kk + khalf];
                afrag[mi] = *(const v2f*)p;
            }
#pragma unroll
            for (int ni = 0; ni < 4; ++ni) {
                int ncol = wn * 64 + ni * 16 + r16;
                v2f b;
                b.x = Bs[(kk + khalf)     * GBN + ncol];
                b.y = Bs[(kk + khalf + 1) * GBN + ncol];
                bfrag[ni] = b;
            }
#pragma unroll
            for (int mi = 0; mi < 2; ++mi)
#pragma unroll
                for (int ni = 0; ni < 4; ++ni)
                    acc[mi][ni] = __builtin_amdgcn_wmma_f32_16x16x4_f32(
                        false, afrag[mi], false, bfrag[ni],
                        (short)0, acc[mi][ni], false, false);
        }
        __syncthreads();
    }

    // ---- epilogue: bias + optional relu, guarded store ----
#pragma unroll
    for (int mi = 0; mi < 2; ++mi) {
#pragma unroll
        for (int ni = 0; ni < 4; ++ni) {
            int col = blockN + wn * 64 + ni * 16 + r16;
            float bv = (col < N) ? bias[col] : 0.f;
#pragma unroll
            for (int r = 0; r < 8; ++r) {
                int row = blockM + wm * 32 + mi * 16 + r + ((lane >> 4) << 3);
                float v = acc[mi][ni][r] + bv;
                if (relu) v = fmaxf(v, 0.f);
                if (row < M && col < N)
                    C[(size_t)row * ldc + col] = v;
            }
        }
    }
}

// ---------------- launcher ----------------
extern "C" void kernel_launch(void* const* d_in, const int* in_sizes, int n_in,
                              void* d_out, int out_size, void* d_ws, size_t ws_size,
                              hipStream_t stream) {
    const float* beta    = (const float*)d_in[0];
    const float* actions = (const float*)d_in[1];
    const float* power   = (const float*)d_in[2];
    const int*   prev    = (const int*)  d_in[3];
    const float* W1      = (const float*)d_in[4];
    const float* b1      = (const float*)d_in[5];
    const float* W2      = (const float*)d_in[6];
    const float* b2      = (const float*)d_in[7];
    const float* W3      = (const float*)d_in[8];
    const float* b3      = (const float*)d_in[9];
    float* out = (float*)d_out;

    float* ws = (float*)d_ws;
    float* tb    = ws + WS_TB;
    int*   tasks = (int*)(ws + WS_TASKS);
    float* best  = ws + WS_BEST;
    int*   neigh = (int*)(ws + WS_NEIGH);
    float* W1p   = ws + WS_W1P;
    float* feat  = ws + WS_FEAT;
    float* X1    = ws + WS_X1;
    float* X2    = ws + WS_X2;

    const int TPB = 256;

    k_total_beta<<<(ROWS * MTASK + TPB - 1) / TPB, TPB, 0, stream>>>(beta, tb);
    k_topk_tasks<<<(ROWS + TPB - 1) / TPB, TPB, 0, stream>>>(tb, tasks);
    k_best<<<(BT * NAG * NAG + TPB - 1) / TPB, TPB, 0, stream>>>(tb, tasks, best);
    k_topk_nb<<<(ROWS + TPB - 1) / TPB, TPB, 0, stream>>>(best, neigh);
    k_w1pad<<<(IN_PAD * HH + TPB - 1) / TPB, TPB, 0, stream>>>(W1, W1p);
    k_features<<<ROWS, TPB, 0, stream>>>(beta, actions, power, prev, tasks, neigh, feat);

    // GEMM1: feat[12800,1664] x W1p[1664,512] + b1, relu -> X1
    {
        dim3 grid(HH / GBN, ROWS / GBM);
        k_gemm_wmma<<<grid, TPB, 0, stream>>>(feat, IN_PAD, W1p, HH, b1,
                                              X1, HH, ROWS, HH, IN_PAD, 1);
    }
    // GEMM2: X1[12800,512] x W2[512,512] + b2, relu -> X2
    {
        dim3 grid(HH / GBN, ROWS / GBM);
        k_gemm_wmma<<<grid, TPB, 0, stream>>>(X1, HH, W2, HH, b2,
                                              X2, HH, ROWS, HH, HH, 1);
    }
    // GEMM3: X2[12800,512] x W3[512,11] + b3 -> out
    {
        dim3 grid((OUT_DIM + GBN - 1) / GBN, ROWS / GBM);
        k_gemm_wmma<<<grid, TPB, 0, stream>>>(X2, HH, W3, OUT_DIM, b3,
                                              out, OUT_DIM, ROWS, OUT_DIM, HH, 0);
    }
}